// DAGEncoder_11613591568817
// MI455X (gfx1250) — compile-verified
//
#include <hip/hip_runtime.h>
#include <hip/hip_bf16.h>

typedef __attribute__((ext_vector_type(2))) float v2f;
typedef __attribute__((ext_vector_type(8))) float v8f;

// ---------------------------------------------------------------------------
// Zero-fill
// ---------------------------------------------------------------------------
__global__ void fill_zero(float* __restrict__ p, int n) {
    int i = blockIdx.x * blockDim.x + threadIdx.x;
    if (i < n) p[i] = 0.0f;
}

// ---------------------------------------------------------------------------
// In/out degree counts:  cdst[dst[e]] += 1,  csrc[src[e]] += 1
// ---------------------------------------------------------------------------
__global__ void degree_count(const int* __restrict__ ei, int E,
                             float* __restrict__ cdst, float* __restrict__ csrc) {
    int e = blockIdx.x * blockDim.x + threadIdx.x;
    if (e < E) {
        atomicAdd(&csrc[ei[e]],     1.0f);   // out-degree (i_idx = src)
        atomicAdd(&cdst[ei[e + E]], 1.0f);   // in-degree  (i_idx = dst)
    }
}

// ---------------------------------------------------------------------------
// Scatter-add of 64-wide node messages over edges:
//   S[iidx[e], c] += Y[jidx[e], c]
// One thread per (edge, channel): a wave covers 32 contiguous channels of one
// edge -> coalesced 128B gather + atomics into one cacheline pair. S (12.8 MB)
// is L2-resident on MI455X (192 MB L2), so the atomics stay on-chip.
// ---------------------------------------------------------------------------
__global__ void scatter_add64(const float* __restrict__ Y,
                              const int* __restrict__ jidx,
                              const int* __restrict__ iidx,
                              float* __restrict__ S, int E) {
    int t = blockIdx.x * blockDim.x + threadIdx.x;
    int e = t >> 6;
    int c = t & 63;
    if (e < E) {
        int j = jidx[e];
        int i = iidx[e];
        atomicAdd(&S[(size_t)i * 64 + c], Y[(size_t)j * 64 + c]);
    }
}

// ---------------------------------------------------------------------------
// Fused GEMM + bias + ReLU using V_WMMA_F32_16X16X4_F32 (full f32 fidelity).
//
//   out[n, ocol0 + h] = relu( sum_k Xcat[n,k] * W[k,h] + bias[h] ),  h in [0,64)
//   Xcat = concat(X0[n, 0:d0], X1[n, 0:Ktot-d0] * (cnt ? 1/max(cnt[n],1) : 1))
//
// Block = 128 threads = 4 waves; each wave computes a 16x64 tile with four
// 16x16 f32 accumulators. The K-loop is split into two WAVE-UNIFORM phases
// (X0 then X1) so the concat never produces divergent control flow, and A
// fragments are prefetched 8 K-steps (one b64 per step) at a time so loadcnt
// drains while the 32 WMMAs of the chunk execute. Rows are clamped (not
// predicated) so the hot loop has no EXEC manipulation at all; the epilogue
// guards stores. W (<= 192x64 = 48 KB) is staged in LDS with (k,k+1) pairs
// adjacent so each B fragment is half of a ds_load_2addr_b64.
//
// Requires d0 % 32 == 0 and (Ktot - d0) % 32 == 0 (true for all call sites:
// d0 = 128, d1 in {0, 64}).
// ---------------------------------------------------------------------------
__global__ __launch_bounds__(128)
void gemm_wmma_relu(const float* __restrict__ X0, int d0,
                    const float* __restrict__ X1,       // may be null
                    const float* __restrict__ cnt,      // may be null (scales X1)
                    const float* __restrict__ W,        // [Ktot, 64]
                    int Ktot,
                    const float* __restrict__ bias,     // may be null, [64]
                    float* __restrict__ out, int ostride, int ocol0, int dup,
                    int Nrows) {
    __shared__ float sW[192 * 64];   // max Ktot = 192

    const int tid = threadIdx.x;

    // Cooperative load of W into LDS, swizzled so (k, k+1) per column are
    // adjacent: element (k,c) -> sW[((k>>1)*64 + c)*2 + (k&1)]
    for (int i = tid; i < Ktot * 64; i += 128) {
        int k = i >> 6;
        int c = i & 63;
        sW[(((k >> 1) << 6) + c) * 2 + (k & 1)] = W[i];
    }
    __syncthreads();

    const int wave = tid >> 5;
    const int lane = tid & 31;
    const int m    = lane & 15;   // A row in tile / B+D col in tile
    const int kSel = lane >> 4;   // 0: holds K,K+1   1: holds K+2,K+3

    const int rowbase = (blockIdx.x * 4 + wave) * 16;
    const int row     = rowbase + m;
    const int crow    = row < Nrows ? row : (Nrows - 1);  // clamp: no divergence

    const float* x0r = X0 + (size_t)crow * d0 + 2 * kSel;
    const float* x1r = X1 ? X1 + (size_t)crow * 64 + 2 * kSel : nullptr;
    float scale1 = 1.0f;
    if (cnt) scale1 = 1.0f / fmaxf(cnt[crow], 1.0f);

    v8f acc[4] = {};

    // ---- Phase 1: X0 columns [0, d0), chunked 8 K-steps at a time ----
#pragma unroll 1
    for (int k0 = 0; k0 < d0; k0 += 32) {
        v2f areg[8];
#pragma unroll
        for (int u = 0; u < 8; ++u)
            areg[u] = *(const v2f*)(x0r + k0 + 4 * u);
#pragma unroll
        for (int u = 0; u < 8; ++u) {
            const int sbase = (((k0 >> 1) + 2 * u + kSel) << 7);
#pragma unroll
            for (int t = 0; t < 4; ++t) {
                v2f b = *(const v2f*)&sW[sbase + ((t << 4) + m) * 2];
                acc[t] = __builtin_amdgcn_wmma_f32_16x16x4_f32(
                    false, areg[u], false, b, (short)0, acc[t], false, false);
            }
        }
    }

    // ---- Phase 2: X1 (aggregated) columns [d0, Ktot), mean fused in ----
    if (x1r) {
#pragma unroll 1
        for (int k0 = d0; k0 < Ktot; k0 += 32) {
            v2f areg[8];
#pragma unroll
            for (int u = 0; u < 8; ++u)
                areg[u] = *(const v2f*)(x1r + (k0 - d0) + 4 * u);
#pragma unroll
            for (int u = 0; u < 8; ++u) {
                v2f a;
                a.x = areg[u].x * scale1;
                a.y = areg[u].y * scale1;
                const int sbase = (((k0 >> 1) + 2 * u + kSel) << 7);
#pragma unroll
                for (int t = 0; t < 4; ++t) {
                    v2f b = *(const v2f*)&sW[sbase + ((t << 4) + m) * 2];
                    acc[t] = __builtin_amdgcn_wmma_f32_16x16x4_f32(
                        false, a, false, b, (short)0, acc[t], false, false);
                }
            }
        }
    }

    // Epilogue: C/D layout -> VGPR r: row = r + 8*(lane>=16), col = lane&15
    const int rr  = kSel * 8;
    const int col = m;
#pragma unroll
    for (int r = 0; r < 8; ++r) {
        const int orow = rowbase + r + rr;
        if (orow < Nrows) {
#pragma unroll
            for (int t = 0; t < 4; ++t) {
                float v = acc[t][r];
                if (bias) v += bias[(t << 4) + col];
                v = fmaxf(v, 0.0f);
                float* p = out + (size_t)orow * ostride + ocol0 + (t << 4) + col;
                *p = v;
                if (dup) p[64] = v;
            }
        }
    }
}

// ---------------------------------------------------------------------------
// Host-side orchestration
// ---------------------------------------------------------------------------
extern "C" void kernel_launch(void* const* d_in, const int* in_sizes, int n_in,
                              void* d_out, int out_size, void* d_ws, size_t ws_size,
                              hipStream_t stream) {
    const float* x    = (const float*)d_in[0];
    const int*   ei   = (const int*)  d_in[1];
    const float* W_u1 = (const float*)d_in[2];
    const float* b_u1 = (const float*)d_in[3];
    const float* U_u1 = (const float*)d_in[4];
    const float* W_d1 = (const float*)d_in[5];
    const float* b_d1 = (const float*)d_in[6];
    const float* U_d1 = (const float*)d_in[7];
    const float* W_u2 = (const float*)d_in[8];
    const float* b_u2 = (const float*)d_in[9];
    const float* U_u2 = (const float*)d_in[10];
    float* out = (float*)d_out;

    const int N = in_sizes[0] / 128;
    const int E = in_sizes[1] / 2;
    const int* src = ei;
    const int* dst = ei + E;

    // Workspace layout (floats): Y[N*64] S[N*64] ha[N*128] hb[N*128] cdst[N] csrc[N]
    float* ws   = (float*)d_ws;
    float* Y    = ws;
    float* S    = Y    + (size_t)N * 64;
    float* ha   = S    + (size_t)N * 64;
    float* hb   = ha   + (size_t)N * 128;
    float* cdst = hb   + (size_t)N * 128;
    float* csrc = cdst + N;

    const int TPB = 256;

    // Degrees (cdst & csrc are contiguous -> single zero pass)
    fill_zero<<<(2 * N + TPB - 1) / TPB, TPB, 0, stream>>>(cdst, 2 * N);
    degree_count<<<(E + TPB - 1) / TPB, TPB, 0, stream>>>(ei, E, cdst, csrc);

    const int gblk  = (N + 63) / 64;
    const int nS    = N * 64;
    const int sgrid = (E * 64 + TPB - 1) / TPB;

    auto conv = [&](const float* Xin, const int* jidx, const int* iidx,
                    const float* cntp, const float* Wm, const float* bm,
                    const float* Um, float* outp, int ocol0, int dupw) {
        // Y = relu(Xin @ Wm + bm)   (per-node message precompute, E/N=16x savings)
        gemm_wmma_relu<<<gblk, 128, 0, stream>>>(Xin, 128, nullptr, nullptr,
                                                 Wm, 128, bm, Y, 64, 0, 0, N);
        fill_zero<<<(nS + TPB - 1) / TPB, TPB, 0, stream>>>(S, nS);
        scatter_add64<<<sgrid, TPB, 0, stream>>>(Y, jidx, iidx, S, E);
        // out = relu([Xin | S/max(cnt,1)] @ Um)
        gemm_wmma_relu<<<gblk, 128, 0, stream>>>(Xin, 128, S, cntp,
                                                 Um, 192, nullptr,
                                                 outp, 128, ocol0, dupw, N);
    };

    // Layer 1: up (src->dst) into ha[:,0:64], down (dst->src) into ha[:,64:128]
    conv(x,  src, dst, cdst, W_u1, b_u1, U_u1, ha,  0, 0);
    conv(x,  dst, src, csrc, W_d1, b_d1, U_d1, ha, 64, 0);
    // Rounds 2..K: both halves use up_conv2 (faithful to source); dup-write
    conv(ha, src, dst, cdst, W_u2, b_u2, U_u2, hb,  0, 1);
    conv(hb, src, dst, cdst, W_u2, b_u2, U_u2, out, 0, 1);
}